// AxonalConnections_80917183857191
// MI455X (gfx1250) — compile-verified
//
#include <hip/hip_runtime.h>
#include <cstdint>
#include <cstddef>

// ---------------------------------------------------------------------------
// AxonalConnections: hub = v1[:,::2,::2]*w_v1_hub + v2[:,::2,::2]*w_v2_hub
//                    out = v1[:,::2,::2]*w_v1_out + v2[:,::2,::2]*w_v2_out
// Bandwidth-bound (AI ~0.2 FLOP/B): optimize the data path, not the ALUs.
//  - weights staged per-block via CDNA5 async global->LDS (ASYNCcnt path)
//  - B128 NT loads of source rows (odd cols share cachelines anyway)
//  - B128 NT stores of both outputs
//  - global_prefetch for next batch iteration
// ---------------------------------------------------------------------------

typedef float v4f __attribute__((ext_vector_type(4)));

#define B_TOT 32
#define SH 1024
#define SW 1024
#define TH 512
#define TW 512
#define ROWS_PER_BLK 4     // target rows per block
#define BATCH_PER_BLK 4    // batches per block
#define NTHREADS 256       // 8 wave32 per block

// CDNA5 async DMA: memory -> LDS, tracked by ASYNCcnt (not LOADcnt).
// vdst = VGPR holding LDS byte address, vaddr = 64-bit global address.
__device__ __forceinline__ void async_copy_b128(uint32_t lds_byte_addr,
                                                const float* gptr) {
  asm volatile("global_load_async_to_lds_b128 %0, %1, off"
               :: "v"(lds_byte_addr), "v"(gptr)
               : "memory");
}

__device__ __forceinline__ void wait_asynccnt0() {
  asm volatile("s_wait_asynccnt 0" ::: "memory");
}

__global__ __launch_bounds__(NTHREADS)
void axonal_connections_kernel(const float* __restrict__ v1,
                               const float* __restrict__ v2,
                               const float* __restrict__ w_v1_hub,
                               const float* __restrict__ w_v2_hub,
                               const float* __restrict__ w_v1_out,
                               const float* __restrict__ w_v2_out,
                               float* __restrict__ hub,
                               float* __restrict__ out) {
  // 4 weight matrices x (4 rows x 512 cols) = 32 KB LDS (of 320 KB/WGP)
  __shared__ float smw[4 * ROWS_PER_BLK * TW];

  const int t  = threadIdx.x;
  const int r0 = blockIdx.x * ROWS_PER_BLK;     // target row tile base
  const int b0 = blockIdx.y * BATCH_PER_BLK;    // batch chunk base

  // ---- async-stage the 4 weight tiles (rows r0..r0+3, full width) --------
  // Each tile is one contiguous 8 KB chunk (row-major, full-width rows).
  // 256 threads x 2 x B128 per matrix = 32 KB total.
  const float* wsrc[4] = { w_v1_hub, w_v2_hub, w_v1_out, w_v2_out };
#pragma unroll
  for (int m = 0; m < 4; ++m) {
    const float* g = wsrc[m] + (size_t)r0 * TW + t * 8;
    uint32_t l = (uint32_t)(uintptr_t)&smw[m * (ROWS_PER_BLK * TW) + t * 8];
    async_copy_b128(l, g);
    async_copy_b128(l + 16u, g + 4);
  }
  wait_asynccnt0();     // this wave's async LDS writes have landed
  __syncthreads();      // all waves' writes visible block-wide

  // ---- per-thread geometry ----------------------------------------------
  const int ry   = t >> 6;            // 0..3  : row within tile
  const int cseg = (t & 63) * 8;      // 0..504: target col base (8 cols/thread)
  const int trow = r0 + ry;           // target row
  const int srow = trow * 2;          // source row (stride-2 gather)
  const int scol = cseg * 2;          // source col base

  // hoist this thread's 8 weights per matrix LDS -> VGPRs (reused 4 batches)
  float wreg[4][8];
#pragma unroll
  for (int m = 0; m < 4; ++m) {
    const float* lw = &smw[m * (ROWS_PER_BLK * TW) + ry * TW + cseg];
#pragma unroll
    for (int j = 0; j < 8; ++j) wreg[m][j] = lw[j];
  }

  const size_t src_base = (size_t)srow * SW + scol;
  const size_t dst_base = (size_t)trow * TW + cseg;

#pragma unroll
  for (int bi = 0; bi < BATCH_PER_BLK; ++bi) {
    const int b = b0 + bi;
    const float* p1 = v1 + (size_t)b * (SH * SW) + src_base;
    const float* p2 = v2 + (size_t)b * (SH * SW) + src_base;

    // prefetch next batch's source lines (global_prefetch_b8)
    if (bi + 1 < BATCH_PER_BLK) {
      __builtin_prefetch(p1 + (size_t)(SH * SW), 0, 0);
      __builtin_prefetch(p2 + (size_t)(SH * SW), 0, 0);
    }

    // 16 consecutive source floats per input = 4 x global_load_b128 (NT:
    // single-use stream, keep L2 for the weights)
    v4f A[4], C[4];
#pragma unroll
    for (int k = 0; k < 4; ++k) {
      A[k] = __builtin_nontemporal_load((const v4f*)(p1 + 4 * k));
      C[k] = __builtin_nontemporal_load((const v4f*)(p2 + 4 * k));
    }

    // stride-2 column gather: take even elements
    float e1[8], e2[8];
#pragma unroll
    for (int k = 0; k < 4; ++k) {
      e1[2 * k]     = A[k].x;  e1[2 * k + 1] = A[k].z;
      e2[2 * k]     = C[k].x;  e2[2 * k + 1] = C[k].z;
    }

    v4f h[2], o[2];
#pragma unroll
    for (int k = 0; k < 2; ++k) {
#pragma unroll
      for (int j = 0; j < 4; ++j) {
        const int i = 4 * k + j;
        h[k][j] = fmaf(e1[i], wreg[0][i], e2[i] * wreg[1][i]);
        o[k][j] = fmaf(e1[i], wreg[2][i], e2[i] * wreg[3][i]);
      }
    }

    float* hp = hub + (size_t)b * (TH * TW) + dst_base;
    float* op = out + (size_t)b * (TH * TW) + dst_base;
    __builtin_nontemporal_store(h[0], (v4f*)hp);
    __builtin_nontemporal_store(h[1], (v4f*)(hp + 4));
    __builtin_nontemporal_store(o[0], (v4f*)op);
    __builtin_nontemporal_store(o[1], (v4f*)(op + 4));
  }
}

extern "C" void kernel_launch(void* const* d_in, const int* in_sizes, int n_in,
                              void* d_out, int out_size, void* d_ws, size_t ws_size,
                              hipStream_t stream) {
  (void)in_sizes; (void)n_in; (void)out_size; (void)d_ws; (void)ws_size;
  const float* v1       = (const float*)d_in[0];
  const float* v2       = (const float*)d_in[1];
  const float* w_v1_hub = (const float*)d_in[2];
  const float* w_v2_hub = (const float*)d_in[3];
  const float* w_v1_out = (const float*)d_in[4];
  const float* w_v2_out = (const float*)d_in[5];

  float* hub = (float*)d_out;                            // first tuple element
  float* out = hub + (size_t)B_TOT * TH * TW;            // second tuple element

  dim3 grid(TH / ROWS_PER_BLK, B_TOT / BATCH_PER_BLK, 1);  // 128 x 8 blocks
  axonal_connections_kernel<<<grid, NTHREADS, 0, stream>>>(
      v1, v2, w_v1_hub, w_v2_hub, w_v1_out, w_v2_out, hub, out);
}